// RobustMeanInterpolate_79980880986183
// MI455X (gfx1250) — compile-verified
//
#include <hip/hip_runtime.h>

// Problem constants (from reference): x is [B,C,H,W] = [4,16,1024,1024] f32.
#define B_ 4
#define C_ 16
#define H_ 1024
#define W_ 1024
#define BC_ (B_ * C_)          // 64 channels
#define NPC_ (H_ * W_)         // 1,048,576 pixels per channel
#define SPLIT_ 32              // reduction splits per channel
#define CHUNK_ (NPC_ / SPLIT_) // 32768 floats per reduce block

#define TILE_ 32
#define HALO_ 2
#define LROWS_ (TILE_ + 2 * HALO_) // 36
#define LW_ 40                     // padded LDS row stride (words)

typedef __attribute__((ext_vector_type(4))) float v4f; // native vector for NT store

// ---------------- CDNA5 async global->LDS helpers ---------------------------
#if defined(__gfx1250__)
typedef __attribute__((address_space(1))) int g_as_int;
typedef __attribute__((address_space(3))) int l_as_int;
#endif

__device__ __forceinline__ void async_load_f32(const float* gsrc, float* ldst) {
#if defined(__gfx1250__) && __has_builtin(__builtin_amdgcn_global_load_async_to_lds_b32)
  __builtin_amdgcn_global_load_async_to_lds_b32(
      (g_as_int*)gsrc, (l_as_int*)ldst, /*offset=*/0, /*cpol=*/0);
#elif defined(__gfx1250__)
  unsigned lofs = (unsigned)(size_t)(__attribute__((address_space(3))) float*)ldst;
  asm volatile("global_load_async_to_lds_b32 %0, %1, off"
               :: "v"(lofs), "v"(gsrc) : "memory");
#else
  *ldst = *gsrc; // host-side / non-gfx1250 fallback
#endif
}

__device__ __forceinline__ void wait_async0() {
#if defined(__gfx1250__) && __has_builtin(__builtin_amdgcn_s_wait_asynccnt)
  __builtin_amdgcn_s_wait_asynccnt(0);
#elif defined(__gfx1250__)
  asm volatile("s_wait_asynccnt 0" ::: "memory");
#endif
}

__device__ __forceinline__ int clampi(int v, int lo, int hi) {
  return v < lo ? lo : (v > hi ? hi : v);
}

// Per-pixel fill: passthrough if valid, else 3x3 valid-mean, else 5x5, else gmean.
__device__ __forceinline__ float fill_pixel(const float* p, float gmean) {
  const float v = p[0];
  if (v == v) return v; // not NaN: passthrough
  float s3 = 0.f, c3 = 0.f;
#pragma unroll
  for (int dy = -1; dy <= 1; ++dy)
#pragma unroll
    for (int dx = -1; dx <= 1; ++dx) {
      float t = p[dy * LW_ + dx];
      bool ok = (t == t);
      s3 += ok ? t : 0.f;
      c3 += ok ? 1.f : 0.f;
    }
  if (c3 > 0.f) return s3 / c3;
  float s5 = 0.f, c5 = 0.f;
#pragma unroll
  for (int dy = -2; dy <= 2; ++dy)
#pragma unroll
    for (int dx = -2; dx <= 2; ++dx) {
      float t = p[dy * LW_ + dx];
      bool ok = (t == t);
      s5 += ok ? t : 0.f;
      c5 += ok ? 1.f : 0.f;
    }
  return (c5 > 0.f) ? (s5 / c5) : gmean;
}

// ---------------- Pass 2: stencil fill (3x3 -> 5x5 -> global mean) ----------
__global__ __launch_bounds__(256)
void fill_kernel(const float* __restrict__ x, const float* __restrict__ means,
                 float* __restrict__ out) {
  __shared__ float tile[LROWS_ * LW_]; // 36 x 40 floats = 5.6 KB

  const int bc = blockIdx.z;
  const int x0 = blockIdx.x * TILE_;
  const int y0 = blockIdx.y * TILE_;
  const float* xc = x + (size_t)bc * NPC_;
  float* oc = out + (size_t)bc * NPC_;
  const int tid  = threadIdx.x;
  const int lane = tid & 31; // lane within wave32
  const int wv   = tid >> 5; // wave id 0..7

  // Stage the clamp-padded (replicate-pad) 36x36 halo tile into LDS via
  // per-lane async global->LDS copies (CDNA5 ASYNCcnt path). Each wave
  // handles whole rows: scalar row clamp, coalesced 128B per async op.
  const int gxa = clampi(x0 - HALO_ + lane, 0, W_ - 1);      // cols 0..31
  const int gxb = clampi(x0 - HALO_ + 32 + lane, 0, W_ - 1); // cols 32..35
#pragma unroll
  for (int k = 0; k < 5; ++k) {
    const int r = wv + k * 8;
    if (r < LROWS_) {
      const int gy = clampi(y0 - HALO_ + r, 0, H_ - 1);
      const float* rowp = xc + (size_t)gy * W_;
      float* ldsrow = &tile[r * LW_];
      async_load_f32(rowp + gxa, ldsrow + lane);
      if (lane < LROWS_ - 32) async_load_f32(rowp + gxb, ldsrow + 32 + lane);
    }
  }
  wait_async0();
  __syncthreads();

  const float gmean = means[bc];
  // Each thread produces 4 consecutive pixels in one row -> one 16B NT store.
  const int row  = tid >> 3;        // 0..31
  const int colq = (tid & 7) * 4;   // 0,4,...,28
  const float* p = &tile[(HALO_ + row) * LW_ + HALO_ + colq];

  v4f res;
  res.x = fill_pixel(p + 0, gmean);
  res.y = fill_pixel(p + 1, gmean);
  res.z = fill_pixel(p + 2, gmean);
  res.w = fill_pixel(p + 3, gmean);

  // Output is never re-read: non-temporal 128-bit store keeps L2 for halos.
  v4f* dst = (v4f*)(oc + (size_t)(y0 + row) * W_ + (x0 + colq));
  __builtin_nontemporal_store(res, dst);
}

// ---------------- Pass 1: per-channel nan-aware partial reduction -----------
__global__ __launch_bounds__(256)
void chan_reduce(const float* __restrict__ x,
                 float* __restrict__ psum, float* __restrict__ pcnt) {
  const int chan  = blockIdx.x / SPLIT_;
  const int split = blockIdx.x % SPLIT_;
  const v4f* p4 = (const v4f*)(x + (size_t)chan * NPC_ + (size_t)split * CHUNK_);

  float s = 0.f, cn = 0.f;
  for (int i = threadIdx.x; i < CHUNK_ / 4; i += 256) {
    v4f v = p4[i];
    bool a = (v.x == v.x), b = (v.y == v.y), c = (v.z == v.z), d = (v.w == v.w);
    s  += (a ? v.x : 0.f) + (b ? v.y : 0.f) + (c ? v.z : 0.f) + (d ? v.w : 0.f);
    cn += (a ? 1.f : 0.f) + (b ? 1.f : 0.f) + (c ? 1.f : 0.f) + (d ? 1.f : 0.f);
  }

  __shared__ float ss[256];
  __shared__ float sc[256];
  ss[threadIdx.x] = s;
  sc[threadIdx.x] = cn;
  __syncthreads();
  for (int o = 128; o > 0; o >>= 1) {
    if (threadIdx.x < o) {
      ss[threadIdx.x] += ss[threadIdx.x + o];
      sc[threadIdx.x] += sc[threadIdx.x + o];
    }
    __syncthreads();
  }
  if (threadIdx.x == 0) {
    psum[blockIdx.x] = ss[0];
    pcnt[blockIdx.x] = sc[0];
  }
}

// ---------------- Pass 1b: fold partials into per-channel means -------------
__global__ void finalize_means(const float* __restrict__ psum,
                               const float* __restrict__ pcnt,
                               float* __restrict__ means) {
  int c = threadIdx.x;
  if (c < BC_) {
    float s = 0.f, cn = 0.f;
    for (int i = 0; i < SPLIT_; ++i) {
      s  += psum[c * SPLIT_ + i];
      cn += pcnt[c * SPLIT_ + i];
    }
    means[c] = s / fmaxf(cn, 1.f);
  }
}

extern "C" void kernel_launch(void* const* d_in, const int* in_sizes, int n_in,
                              void* d_out, int out_size, void* d_ws, size_t ws_size,
                              hipStream_t stream) {
  const float* x = (const float*)d_in[0];
  float* out = (float*)d_out;

  float* psum  = (float*)d_ws;            // BC_*SPLIT_ floats
  float* pcnt  = psum + BC_ * SPLIT_;     // BC_*SPLIT_ floats
  float* means = pcnt + BC_ * SPLIT_;     // BC_ floats

  chan_reduce<<<BC_ * SPLIT_, 256, 0, stream>>>(x, psum, pcnt);
  finalize_means<<<1, 64, 0, stream>>>(psum, pcnt, means);

  dim3 grid(W_ / TILE_, H_ / TILE_, BC_);
  fill_kernel<<<grid, 256, 0, stream>>>(x, means, out);
}